// GraphDecoder_32916629356848
// MI455X (gfx1250) — compile-verified
//
#include <hip/hip_runtime.h>
#include <hip/hip_bf16.h>
#include <math.h>

// MI455X / gfx1250: wave32, WMMA f32 16x16x4 for exact fp32 matrix math.

#define BN_EPS 1e-5f

typedef float v2f __attribute__((ext_vector_type(2)));
typedef float v8f __attribute__((ext_vector_type(8)));

__device__ __forceinline__ v8f wmma4(v2f a, v2f b, v8f c) {
  // D = A(16x4) x B(4x16) + C(16x16), all fp32
  return __builtin_amdgcn_wmma_f32_16x16x4_f32(false, a, false, b,
                                               (short)0, c, false, false);
}

// ---------------------------------------------------------------------------
// Generic C = A @ W.T with fused BN / bias / ReLU epilogue.
// One wave computes one 16x16 output tile. 4 waves per block.
//   out[m,n] = act( (sum_k A[m,k]*W[n, wcol+k]) * s[n] + t[n] )
//   s[n] = g[n]/sqrt(var[n]+eps)          (1 if g==null)
//   t[n] = (bias[n]-mu[n])*s[n] + be[n]   (null pointers treated as 0)
// ---------------------------------------------------------------------------
__global__ __launch_bounds__(128)
void gemm_bn_kernel(const float* __restrict__ A, int lda,
                    const float* __restrict__ W, int ldw, int wcol,
                    float* __restrict__ Out, int ldo,
                    int K, int nTilesN,
                    const float* __restrict__ g,
                    const float* __restrict__ be,
                    const float* __restrict__ mu,
                    const float* __restrict__ var,
                    const float* __restrict__ bias,
                    int relu)
{
  int lane = threadIdx.x & 31;
  int wave = threadIdx.x >> 5;
  int tile = blockIdx.x * 4 + wave;
  int m0 = (tile / nTilesN) << 4;
  int n0 = (tile % nTilesN) << 4;

  int r16  = lane & 15;        // A: row M;  B: col N
  int hi16 = lane >> 4;        // lane half selects K pair
  int kh   = hi16 << 1;        // 0 or 2

  const float* Ap = A + (size_t)(m0 + r16) * lda + kh;
  const float* Wp = W + (size_t)(n0 + r16) * ldw + wcol + kh;

  v8f acc = {};
#pragma unroll 8
  for (int k = 0; k < K; k += 4) {
    v2f a = *(const v2f*)(Ap + k);   // A[m][k+kh], A[m][k+kh+1]
    v2f b = *(const v2f*)(Wp + k);   // B[k+kh][n] = W[n][k+kh], next
    acc = wmma4(a, b, acc);
  }

  int n = n0 + r16;
  float s = g ? (g[n] / sqrtf(var[n] + BN_EPS)) : 1.0f;
  float t = (bias ? bias[n] : 0.0f) - (mu ? mu[n] : 0.0f);
  t = t * s + (be ? be[n] : 0.0f);

#pragma unroll
  for (int v = 0; v < 8; ++v) {
    int m = m0 + v + (hi16 << 3);            // C/D layout: M = v + 8*(lane/16)
    float val = acc[v] * s + t;
    if (relu) val = fmaxf(val, 0.0f);
    Out[(size_t)m * ldo + n] = val;
  }
}

// ---------------------------------------------------------------------------
// Fused edge MLP. Block = 8 waves (256 thr), one (i, j0) tile of 64 edges.
//   eh  = relu(hi'[i] + hj'[j])                      (LDS, 64x512, stride 516)
//   eh2 = relu((eh @ eW2.T)*s2 + t2)                 (LDS, 64x512, stride 516)
//   out = eh2 @ eW3.T + eb3  -> logits (o=0), features (o=1..32)
// ---------------------------------------------------------------------------
#define LDST 516   // 512 + 4 pad -> conflict-free A-fragment ds_load_b64

__global__ __launch_bounds__(256)
void edge_fused_kernel(const float* __restrict__ hiP,   // hi' [512,512]
                       const float* __restrict__ hjP,   // hj' [512,512]
                       const float* __restrict__ eW2,   // [512,512]
                       const float* __restrict__ eb2,
                       const float* __restrict__ g2,
                       const float* __restrict__ be2,
                       const float* __restrict__ mu2,
                       const float* __restrict__ var2,
                       const float* __restrict__ eW3,   // [33,512]
                       const float* __restrict__ eb3,   // [33]
                       float* __restrict__ logits,      // [512*512]
                       float* __restrict__ feats)       // [512*512*32]
{
  extern __shared__ float lds[];
  float* eh  = lds;               // 64 x LDST
  float* eh2 = lds + 64 * LDST;   // 64 x LDST

  int i  = blockIdx.x >> 3;
  int j0 = (blockIdx.x & 7) << 6;

  // ---- phase 0: build eh tile in LDS (float4 vectorized) ----
  const float4* hiv = (const float4*)(hiP + (size_t)i * 512);
  for (int idx = threadIdx.x; idx < 64 * 128; idx += 256) {
    int m = idx >> 7, kq = idx & 127;
    float4 a = hiv[kq];
    float4 b = ((const float4*)(hjP + (size_t)(j0 + m) * 512))[kq];
    float4 r;
    r.x = fmaxf(a.x + b.x, 0.0f);
    r.y = fmaxf(a.y + b.y, 0.0f);
    r.z = fmaxf(a.z + b.z, 0.0f);
    r.w = fmaxf(a.w + b.w, 0.0f);
    *(float4*)(eh + m * LDST + (kq << 2)) = r;
  }
  __syncthreads();

  int lane = threadIdx.x & 31;
  int wave = threadIdx.x >> 5;
  int r16  = lane & 15;
  int hi16 = lane >> 4;
  int kh   = hi16 << 1;

  // ---- phase 1: eh2 = relu(bn2(eh @ eW2.T + eb2)) ----
  for (int nt = wave * 4; nt < wave * 4 + 4; ++nt) {
    int n = nt * 16 + r16;
    const float* wp = eW2 + (size_t)n * 512 + kh;
    const float* ap = eh + r16 * LDST + kh;
    v8f c0 = {}, c1 = {}, c2 = {}, c3 = {};
#pragma unroll 4
    for (int k = 0; k < 512; k += 4) {
      v2f b  = *(const v2f*)(wp + k);
      v2f a0 = *(const v2f*)(ap + k);
      v2f a1 = *(const v2f*)(ap + 16 * LDST + k);
      v2f a2 = *(const v2f*)(ap + 32 * LDST + k);
      v2f a3 = *(const v2f*)(ap + 48 * LDST + k);
      c0 = wmma4(a0, b, c0);
      c1 = wmma4(a1, b, c1);
      c2 = wmma4(a2, b, c2);
      c3 = wmma4(a3, b, c3);
    }
    float s = g2[n] / sqrtf(var2[n] + BN_EPS);
    float t = (eb2[n] - mu2[n]) * s + be2[n];
#pragma unroll
    for (int v = 0; v < 8; ++v) {
      int row = v + (hi16 << 3);
      eh2[(row     ) * LDST + n] = fmaxf(c0[v] * s + t, 0.0f);
      eh2[(row + 16) * LDST + n] = fmaxf(c1[v] * s + t, 0.0f);
      eh2[(row + 32) * LDST + n] = fmaxf(c2[v] * s + t, 0.0f);
      eh2[(row + 48) * LDST + n] = fmaxf(c3[v] * s + t, 0.0f);
    }
  }
  __syncthreads();

  // ---- phase 2: out = eh2 @ eW3.T + eb3 (33 cols -> 3 padded n-tiles) ----
  for (int t = wave; t < 12; t += 8) {
    int mt = t / 3, ot = t - mt * 3;
    int o  = ot * 16 + r16;
    bool ok = (o < 33);
    int oc = ok ? o : 32;                       // clamp: loads stay in-bounds
    const float* wp = eW3 + (size_t)oc * 512 + kh;
    const float* ap = eh2 + (mt * 16 + r16) * LDST + kh;
    v8f c = {};
#pragma unroll 4
    for (int k = 0; k < 512; k += 4) {
      v2f b = *(const v2f*)(wp + k);
      if (!ok) { b.x = 0.0f; b.y = 0.0f; }      // lane mask, EXEC stays full
      v2f a = *(const v2f*)(ap + k);
      c = wmma4(a, b, c);
    }
    if (ok) {
      float bias = eb3[o];
#pragma unroll
      for (int v = 0; v < 8; ++v) {
        int m = mt * 16 + v + (hi16 << 3);
        size_t edge = (size_t)i * 512 + (j0 + m);
        float val = c[v] + bias;
        if (o == 0) logits[edge] = val;
        else        feats[edge * 32 + (o - 1)] = val;
      }
    }
  }
}

// ---------------------------------------------------------------------------
extern "C" void kernel_launch(void* const* d_in, const int* in_sizes, int n_in,
                              void* d_out, int out_size, void* d_ws, size_t ws_size,
                              hipStream_t stream) {
  const float* X        = (const float*)d_in[0];   // [512,256]
  const float* nW1      = (const float*)d_in[1];   // [512,256]
  const float* nb1      = (const float*)d_in[2];
  const float* n_gamma1 = (const float*)d_in[3];
  const float* n_beta1  = (const float*)d_in[4];
  const float* n_mean1  = (const float*)d_in[5];
  const float* n_var1   = (const float*)d_in[6];
  const float* nW2      = (const float*)d_in[7];   // [128,512]
  const float* nb2      = (const float*)d_in[8];
  const float* eW1      = (const float*)d_in[9];   // [512,512]
  const float* eb1      = (const float*)d_in[10];
  const float* e_gamma1 = (const float*)d_in[11];
  const float* e_beta1  = (const float*)d_in[12];
  const float* e_mean1  = (const float*)d_in[13];
  const float* e_var1   = (const float*)d_in[14];
  const float* eW2      = (const float*)d_in[15];  // [512,512]
  const float* eb2      = (const float*)d_in[16];
  const float* e_gamma2 = (const float*)d_in[17];
  const float* e_beta2  = (const float*)d_in[18];
  const float* e_mean2  = (const float*)d_in[19];
  const float* e_var2   = (const float*)d_in[20];
  const float* eW3      = (const float*)d_in[21];  // [33,512]
  const float* eb3      = (const float*)d_in[22];

  float* out    = (float*)d_out;
  float* nodeF  = out;                       // [512,128]
  float* logits = out + 512 * 128;           // [512,512]
  float* feats  = logits + 512 * 512;        // [512,512,32]

  float* ws    = (float*)d_ws;
  float* ws_hi = ws;                         // hi' [512,512]
  float* ws_hj = ws + 512 * 512;             // hj' [512,512]
  float* ws_h1 = ws + 2 * 512 * 512;         // h1  [512,512]

  dim3 blk(128);
  // hi' = (X @ Wa.T) * s1                 (Wa = eW1[:, :256])
  gemm_bn_kernel<<<256, blk, 0, stream>>>(X, 256, eW1, 512, 0, ws_hi, 512,
      256, 32, e_gamma1, nullptr, nullptr, e_var1, nullptr, 0);
  // hj' = (X @ Wb.T) * s1 + (eb1-mean)*s1 + beta   (Wb = eW1[:, 256:])
  gemm_bn_kernel<<<256, blk, 0, stream>>>(X, 256, eW1, 512, 256, ws_hj, 512,
      256, 32, e_gamma1, e_beta1, e_mean1, e_var1, eb1, 0);
  // h1 = relu(bn1(X @ nW1.T + nb1))
  gemm_bn_kernel<<<256, blk, 0, stream>>>(X, 256, nW1, 256, 0, ws_h1, 512,
      256, 32, n_gamma1, n_beta1, n_mean1, n_var1, nb1, 1);
  // node_features = h1 @ nW2.T + nb2      (512x128, 32x8 tiles)
  gemm_bn_kernel<<<64, blk, 0, stream>>>(ws_h1, 512, nW2, 512, 0, nodeF, 128,
      512, 8, nullptr, nullptr, nullptr, nullptr, nb2, 0);

  // fused edge MLP: 512 i-rows x 8 j-tiles of 64 edges
  size_t ldsBytes = (size_t)2 * 64 * LDST * sizeof(float);  // ~258 KB
  edge_fused_kernel<<<4096, 256, ldsBytes, stream>>>(
      ws_hi, ws_hj, eW2, eb2, e_gamma2, e_beta2, e_mean2, e_var2,
      eW3, eb3, logits, feats);
}